// GCN_10222022164775
// MI455X (gfx1250) — compile-verified
//
#include <hip/hip_runtime.h>

typedef __attribute__((ext_vector_type(2))) float v2f;
typedef __attribute__((ext_vector_type(8))) float v8f;

#define N_NODES 100000
#define N_EDGES 800000
#define FEAT    128
#define NGRAPH  512
#define NCLS    10

__device__ __forceinline__ void atomAddF(float* p, float v) {
  unsafeAtomicAdd(p, v);   // native global_atomic_add_f32 on gfx1250
}

// ---------------- fill ----------------
__global__ void k_fill(float* __restrict__ p, float v, long n) {
  long i = (long)blockIdx.x * blockDim.x + threadIdx.x;
  if (i < n) p[i] = v;
}

// ---------------- degree (targets incl. self-loops: deg pre-filled with 1) ----
__global__ void k_deg(const int* __restrict__ col, float* __restrict__ deg) {
  long e = (long)blockIdx.x * blockDim.x + threadIdx.x;
  if (e < N_EDGES) atomAddF(&deg[col[e]], 1.0f);
}

// ---------------- deg -> dinv in place ----------------
__global__ void k_dinv(float* __restrict__ deg) {
  long i = (long)blockIdx.x * blockDim.x + threadIdx.x;
  if (i < N_NODES) {
    float d = deg[i];
    deg[i] = (d > 0.0f) ? rsqrtf(fmaxf(d, 1.0f)) : 0.0f;
  }
}

// ---------------- H = X * W  via V_WMMA_F32_16X16X4_F32 ----------------
// X:[nrows,128] row-major, W:[128,128] row-major, Hout:[nrows,128]
// Block = 256 threads = 8 waves; each wave computes one 16-row strip x 128 cols.
__global__ __launch_bounds__(256) void k_gemm(const float* __restrict__ X,
                                              const float* __restrict__ W,
                                              float* __restrict__ Hout,
                                              int nrows) {
  __shared__ float sW[FEAT * FEAT];   // 64 KB (<= 320 KB/WGP on CDNA5)
  {
    const float4* src = (const float4*)W;
    float4* dst = (float4*)sW;
    for (int i = threadIdx.x; i < FEAT * FEAT / 4; i += blockDim.x) dst[i] = src[i];
  }
  __syncthreads();

  const int wave  = threadIdx.x >> 5;
  const int lane  = threadIdx.x & 31;
  const int strip = blockIdx.x * 8 + wave;
  const int row0  = strip * 16;
  if (row0 >= nrows) return;          // wave-uniform: EXEC stays all-ones for WMMA

  const int m    = lane & 15;         // A row within strip (both lane halves)
  const int koff = (lane >> 4) << 1;  // lanes 16-31 hold K+2,K+3

  v8f acc[8];
#pragma unroll
  for (int n = 0; n < 8; ++n) acc[n] = (v8f){0, 0, 0, 0, 0, 0, 0, 0};

  const float* xrow = X + (size_t)(row0 + m) * FEAT + koff;
  const int ncol = lane & 15;

  for (int k = 0; k < FEAT; k += 4) {
    v2f a;
    a.x = xrow[k];            // A[m][k+koff]
    a.y = xrow[k + 1];        // A[m][k+koff+1]  (merges to b64 load)
    const int kb = k + koff;
#pragma unroll
    for (int n = 0; n < 8; ++n) {
      v2f b;
      b.x = sW[kb * FEAT + n * 16 + ncol];        // B[kb][n0+ncol]
      b.y = sW[(kb + 1) * FEAT + n * 16 + ncol];  // B[kb+1][n0+ncol]
      acc[n] = __builtin_amdgcn_wmma_f32_16x16x4_f32(
          false, a, false, b, (short)0, acc[n], false, false);
    }
  }

  // C/D layout: VGPR r -> row r (lanes 0-15) / row r+8 (lanes 16-31)
  const int rowsel = (lane >> 4) * 8;
#pragma unroll
  for (int n = 0; n < 8; ++n) {
#pragma unroll
    for (int r = 0; r < 8; ++r) {
      Hout[(size_t)(row0 + rowsel + r) * FEAT + n * 16 + ncol] = acc[n][r];
    }
  }
}

// ---------------- edge aggregation: out[col] += dinv[row]*dinv[col]*h[row] ----
// one wave per edge; lane handles 4 consecutive features
__global__ __launch_bounds__(256) void k_agg(const float* __restrict__ h,
                                             const int* __restrict__ row,
                                             const int* __restrict__ col,
                                             const float* __restrict__ dinv,
                                             float* __restrict__ out) {
  const int wave = threadIdx.x >> 5;
  const int lane = threadIdx.x & 31;
  const long e = (long)blockIdx.x * 8 + wave;
  if (e >= N_EDGES) return;
  const int r = row[e];
  const int c = col[e];
  const float nrm = dinv[r] * dinv[c];
  const float4 v = ((const float4*)(h + (size_t)r * FEAT))[lane];
  float* op = out + (size_t)c * FEAT + lane * 4;
  atomAddF(op + 0, nrm * v.x);
  atomAddF(op + 1, nrm * v.y);
  atomAddF(op + 2, nrm * v.z);
  atomAddF(op + 3, nrm * v.w);
}

// ---------------- epilogue: out += dinv^2 * h (self-loop) + bias; optional ReLU
__global__ void k_epilogue(float* __restrict__ out, const float* __restrict__ h,
                           const float* __restrict__ dinv,
                           const float* __restrict__ bias, int relu) {
  long idx = (long)blockIdx.x * blockDim.x + threadIdx.x;  // float4 index
  if (idx >= (long)N_NODES * (FEAT / 4)) return;
  const int node = (int)(idx >> 5);        // FEAT/4 == 32
  const int f4   = (int)(idx & 31);
  const float d = dinv[node];
  const float s = d * d;
  float4 hv = ((const float4*)h)[idx];
  float4 ov = ((float4*)out)[idx];
  float4 bv = ((const float4*)bias)[f4];
  float4 rv;
  rv.x = ov.x + s * hv.x + bv.x;
  rv.y = ov.y + s * hv.y + bv.y;
  rv.z = ov.z + s * hv.z + bv.z;
  rv.w = ov.w + s * hv.w + bv.w;
  if (relu) {
    rv.x = fmaxf(rv.x, 0.0f);
    rv.y = fmaxf(rv.y, 0.0f);
    rv.z = fmaxf(rv.z, 0.0f);
    rv.w = fmaxf(rv.w, 0.0f);
  }
  ((float4*)out)[idx] = rv;
}

// ---------------- pooling accumulation ----------------
__global__ __launch_bounds__(256) void k_pool(const float* __restrict__ a,
                                              const int* __restrict__ batch,
                                              float* __restrict__ sums,
                                              float* __restrict__ cnts) {
  const int wave = threadIdx.x >> 5;
  const int lane = threadIdx.x & 31;
  const long i = (long)blockIdx.x * 8 + wave;
  if (i >= N_NODES) return;
  const int g = batch[i];
  const float4 v = ((const float4*)(a + (size_t)i * FEAT))[lane];
  float* sp = sums + (size_t)g * FEAT + lane * 4;
  atomAddF(sp + 0, v.x);
  atomAddF(sp + 1, v.y);
  atomAddF(sp + 2, v.z);
  atomAddF(sp + 3, v.w);
  if (lane == 0) atomAddF(&cnts[g], 1.0f);
}

// ---------------- final: out[g][c] = (sums[g]/max(cnt,1)) . Wl[:,c] + bl[c] ----
__global__ void k_final(const float* __restrict__ sums, const float* __restrict__ cnts,
                        const float* __restrict__ Wl, const float* __restrict__ bl,
                        float* __restrict__ out) {
  const int g = blockIdx.x;
  const int c = threadIdx.x;
  if (c >= NCLS) return;
  const float inv = 1.0f / fmaxf(cnts[g], 1.0f);
  float acc = bl[c];
  for (int k = 0; k < FEAT; ++k)
    acc += sums[(size_t)g * FEAT + k] * inv * Wl[k * NCLS + c];
  out[(size_t)g * NCLS + c] = acc;
}

extern "C" void kernel_launch(void* const* d_in, const int* in_sizes, int n_in,
                              void* d_out, int out_size, void* d_ws, size_t ws_size,
                              hipStream_t stream) {
  const float* x     = (const float*)d_in[0];
  const int*   ei    = (const int*)d_in[1];
  const int*   row   = ei;             // edge_index[0] = sources
  const int*   col   = ei + N_EDGES;   // edge_index[1] = targets
  const int*   batch = (const int*)d_in[2];
  const float* W1 = (const float*)d_in[3];
  const float* b1 = (const float*)d_in[4];
  const float* W2 = (const float*)d_in[5];
  const float* b2 = (const float*)d_in[6];
  const float* W3 = (const float*)d_in[7];
  const float* b3 = (const float*)d_in[8];
  const float* Wl = (const float*)d_in[9];
  const float* bl = (const float*)d_in[10];
  float* outp = (float*)d_out;

  char* base = (char*)d_ws;
  const size_t featBytes = (size_t)N_NODES * FEAT * sizeof(float);  // 51.2 MB
  float* deg  = (float*)(base);                       // N floats (becomes dinv)
  float* hbuf = (float*)(base + (1 << 20));
  float* abuf = (float*)(base + (1 << 20) + featBytes);
  float* sums = (float*)(base + (1 << 20) + 2 * featBytes);
  float* cnts = sums + (size_t)NGRAPH * FEAT;

  const long nElem  = (long)N_NODES * FEAT;      // 12.8M
  const long nVec4  = nElem / 4;                 // 3.2M
  const int  TB     = 256;
  const int  gNode  = (N_NODES + TB - 1) / TB;
  const int  gElem  = (int)((nElem + TB - 1) / TB);
  const int  gVec4  = (int)((nVec4 + TB - 1) / TB);
  const int  gEdge1 = (N_EDGES + TB - 1) / TB;
  const int  gGemm  = (N_NODES / 16 + 8 - 1) / 8;    // waves of 16-row strips
  const int  gAgg   = (N_EDGES + 8 - 1) / 8;         // 1 wave per edge
  const int  gPool  = (N_NODES + 8 - 1) / 8;

  // normalization: deg over targets (self-loops contribute the initial 1)
  k_fill<<<gNode, TB, 0, stream>>>(deg, 1.0f, N_NODES);
  k_deg<<<gEdge1, TB, 0, stream>>>(col, deg);
  k_dinv<<<gNode, TB, 0, stream>>>(deg);   // deg now holds dinv

  // layer 1
  k_gemm<<<gGemm, TB, 0, stream>>>(x, W1, hbuf, N_NODES);
  k_fill<<<gElem, TB, 0, stream>>>(abuf, 0.0f, nElem);
  k_agg<<<gAgg, TB, 0, stream>>>(hbuf, row, col, deg, abuf);
  k_epilogue<<<gVec4, TB, 0, stream>>>(abuf, hbuf, deg, b1, 1);

  // layer 2
  k_gemm<<<gGemm, TB, 0, stream>>>(abuf, W2, hbuf, N_NODES);
  k_fill<<<gElem, TB, 0, stream>>>(abuf, 0.0f, nElem);
  k_agg<<<gAgg, TB, 0, stream>>>(hbuf, row, col, deg, abuf);
  k_epilogue<<<gVec4, TB, 0, stream>>>(abuf, hbuf, deg, b2, 1);

  // layer 3 (no ReLU)
  k_gemm<<<gGemm, TB, 0, stream>>>(abuf, W3, hbuf, N_NODES);
  k_fill<<<gElem, TB, 0, stream>>>(abuf, 0.0f, nElem);
  k_agg<<<gAgg, TB, 0, stream>>>(hbuf, row, col, deg, abuf);
  k_epilogue<<<gVec4, TB, 0, stream>>>(abuf, hbuf, deg, b3, 0);

  // global mean pool + classifier
  const long poolN = (long)NGRAPH * FEAT + NGRAPH;
  k_fill<<<(int)((poolN + TB - 1) / TB), TB, 0, stream>>>(sums, 0.0f, poolN);
  k_pool<<<gPool, TB, 0, stream>>>(abuf, batch, sums, cnts);
  k_final<<<NGRAPH, 32, 0, stream>>>(sums, cnts, Wl, bl, outp);
}